// HardSession_17635135717525
// MI455X (gfx1250) — compile-verified
//
#include <hip/hip_runtime.h>
#include <math.h>

#define N_NODES 20000
#define N_EDGES 320000
#define DIM     128
#define HEADS   8
#define FEATS   128
#define HF      (HEADS * FEATS)   // 1024
#define NEG_SLOPE 0.2f

typedef __attribute__((ext_vector_type(2))) float v2f;
typedef __attribute__((ext_vector_type(4))) float v4f;
typedef __attribute__((ext_vector_type(8))) float v8f;

// ---------------------------------------------------------------------------
// 1) Projection GEMM: fs[n][o] = sum_k feat[n][k] * W[o][k]
//    One wave32 per 16x16 output tile, V_WMMA_F32_16X16X4_F32, K-loop of 32.
//    A-lane layout (ISA 7.12.2, 32-bit A 16x4): lane<16 -> M=lane, K={0,1};
//    lane>=16 -> M=lane-16, K={2,3}.  B mirrors with N across lanes.
// ---------------------------------------------------------------------------
__global__ __launch_bounds__(256) void k_proj(const float* __restrict__ feat,
                                              const float* __restrict__ W,
                                              float* __restrict__ fs) {
    const int wave = threadIdx.x >> 5;
    const int lane = threadIdx.x & 31;
    const int tile = blockIdx.x * 8 + wave;          // 1250*64 = 80000 tiles
    const int colTile = tile & 63;                   // 64 tiles across HF=1024
    const int rowTile = tile >> 6;                   // 1250 tiles down N
    const int row0 = rowTile * 16;
    const int col0 = colTile * 16;
    const int lm = lane & 15;                        // M for A, N for B
    const int kh = lane >> 4;                        // selects K pair {0,1} vs {2,3}

    const float* arow = feat + (size_t)(row0 + lm) * DIM + 2 * kh;
    const float* brow = W    + (size_t)(col0 + lm) * DIM + 2 * kh;

    v8f c = {};
#pragma unroll 4
    for (int k0 = 0; k0 < DIM; k0 += 4) {
        v2f a = *(const v2f*)(arow + k0);
        v2f b = *(const v2f*)(brow + k0);
        c = __builtin_amdgcn_wmma_f32_16x16x4_f32(false, a, false, b,
                                                  (short)0, c, false, false);
    }
    // C/D layout: element v of lane -> D[M = v + 8*kh][N = lm]
#pragma unroll
    for (int v = 0; v < 8; ++v) {
        fs[(size_t)(row0 + v + 8 * kh) * HF + col0 + lm] = c[v];
    }
}

// ---------------------------------------------------------------------------
// 2) Per-(node,head) attention logits: el = <fs[n,h,:], attn_l[h,:]>, er like.
//    One wave per (node,head); lane covers 4 feats; shuffle reduction.
// ---------------------------------------------------------------------------
__global__ __launch_bounds__(256) void k_elr(const float* __restrict__ fs,
                                             const float* __restrict__ attn_l,
                                             const float* __restrict__ attn_r,
                                             float* __restrict__ el,
                                             float* __restrict__ er) {
    const int wave = threadIdx.x >> 5;
    const int lane = threadIdx.x & 31;
    const int idx  = blockIdx.x * 8 + wave;          // node*HEADS + h
    if (idx >= N_NODES * HEADS) return;
    const int n = idx / HEADS;
    const int h = idx - n * HEADS;

    v4f x  = *(const v4f*)(fs + (size_t)n * HF + h * FEATS + lane * 4);
    v4f al = *(const v4f*)(attn_l + h * FEATS + lane * 4);
    v4f ar = *(const v4f*)(attn_r + h * FEATS + lane * 4);
    float sl = x.x * al.x + x.y * al.y + x.z * al.z + x.w * al.w;
    float sr = x.x * ar.x + x.y * ar.y + x.z * ar.z + x.w * ar.w;
#pragma unroll
    for (int off = 16; off > 0; off >>= 1) {
        sl += __shfl_down(sl, off, 32);
        sr += __shfl_down(sr, off, 32);
    }
    if (lane == 0) { el[idx] = sl; er[idx] = sr; }
}

// ---------------------------------------------------------------------------
// 3) Init: out = residual + bias;  emax = -inf; esum = 0.
// ---------------------------------------------------------------------------
__global__ void k_init(const float* __restrict__ feat,
                       const float* __restrict__ bias,
                       float* __restrict__ out,
                       float* __restrict__ emax,
                       float* __restrict__ esum) {
    const int i = blockIdx.x * blockDim.x + threadIdx.x;
    if (i < N_NODES * HEADS) {
        emax[i] = __int_as_float(0xFF800000);  // -inf
        esum[i] = 0.0f;
    }
    if (i < N_NODES * HF) {
        const int n = i / HF;
        const int r = i - n * HF;              // h*F + f
        const int f = r & (FEATS - 1);
        out[i] = feat[(size_t)n * DIM + f] + bias[r];
    }
}

// Monotonic float atomic max via signed/unsigned integer atomics.
__device__ __forceinline__ void atomicMaxFloat(float* addr, float val) {
    if (val >= 0.0f)
        atomicMax((int*)addr, __float_as_int(val));
    else
        atomicMin((unsigned int*)addr, __float_as_uint(val));
}

// ---------------------------------------------------------------------------
// 4) Edge logits + leaky relu + per-dst max
// ---------------------------------------------------------------------------
__global__ void k_edge_logit(const int* __restrict__ src,
                             const int* __restrict__ dst,
                             const float* __restrict__ el,
                             const float* __restrict__ er,
                             float* __restrict__ e,
                             float* __restrict__ emax) {
    const int i = blockIdx.x * blockDim.x + threadIdx.x;
    if (i >= N_EDGES * HEADS) return;
    const int eid = i / HEADS;
    const int h   = i - eid * HEADS;
    const int s = src[eid];
    const int d = dst[eid];
    float v = el[s * HEADS + h] + er[d * HEADS + h];
    v = v > 0.0f ? v : NEG_SLOPE * v;
    e[i] = v;
    atomicMaxFloat(&emax[d * HEADS + h], v);
}

// ---------------------------------------------------------------------------
// 5) Edge exp + per-dst sum (in-place e -> exp(e - emax[dst]))
// ---------------------------------------------------------------------------
__global__ void k_edge_exp(const int* __restrict__ dst,
                           float* __restrict__ e,
                           const float* __restrict__ emax,
                           float* __restrict__ esum) {
    const int i = blockIdx.x * blockDim.x + threadIdx.x;
    if (i >= N_EDGES * HEADS) return;
    const int eid = i / HEADS;
    const int h   = i - eid * HEADS;
    const int d = dst[eid];
    const float v = expf(e[i] - emax[d * HEADS + h]);
    e[i] = v;
    atomicAdd(&esum[d * HEADS + h], v);
}

// ---------------------------------------------------------------------------
// 6) Scatter: out[dst,h,:] += fs[src,h,:] * (ee / esum[dst,h])
//    One thread per (edge, head, 4-feat chunk); f fastest within wave so the
//    fs gather and atomic adds are contiguous across 32 lanes (L2-friendly).
// ---------------------------------------------------------------------------
__global__ void k_scatter(const int* __restrict__ src,
                          const int* __restrict__ dst,
                          const float* __restrict__ fs,
                          const float* __restrict__ e,
                          const float* __restrict__ esum,
                          float* __restrict__ out) {
    const unsigned int i = blockIdx.x * blockDim.x + threadIdx.x;
    if (i >= (unsigned int)N_EDGES * HEADS * 32u) return;
    const unsigned int fq = i & 31u;       // 4-feat chunk index
    const unsigned int t  = i >> 5;
    const int h   = (int)(t % HEADS);
    const int eid = (int)(t / HEADS);
    const int s = src[eid];
    const int d = dst[eid];
    const float a = e[eid * HEADS + h] / esum[d * HEADS + h];
    const v4f m = *(const v4f*)(fs + (size_t)s * HF + h * FEATS + fq * 4);
    float* op = out + (size_t)d * HF + h * FEATS + fq * 4;
    atomicAdd(op + 0, m.x * a);
    atomicAdd(op + 1, m.y * a);
    atomicAdd(op + 2, m.z * a);
    atomicAdd(op + 3, m.w * a);
}

// ---------------------------------------------------------------------------
extern "C" void kernel_launch(void* const* d_in, const int* in_sizes, int n_in,
                              void* d_out, int out_size, void* d_ws, size_t ws_size,
                              hipStream_t stream) {
    const float* feat   = (const float*)d_in[0];
    const int*   src    = (const int*)  d_in[1];
    const int*   dst    = (const int*)  d_in[2];
    const float* W_fc   = (const float*)d_in[3];
    const float* attn_l = (const float*)d_in[4];
    const float* attn_r = (const float*)d_in[5];
    const float* bias   = (const float*)d_in[6];
    float* out = (float*)d_out;

    // Workspace layout (floats)
    float* fs   = (float*)d_ws;                         // N*HF   = 20,480,000
    float* el   = fs   + (size_t)N_NODES * HF;          // N*H    = 160,000
    float* er   = el   + (size_t)N_NODES * HEADS;
    float* emax = er   + (size_t)N_NODES * HEADS;
    float* esum = emax + (size_t)N_NODES * HEADS;
    float* e    = esum + (size_t)N_NODES * HEADS;       // E*H    = 2,560,000

    // 1) projection GEMM (WMMA f32): 80000 tiles, 8 waves/block
    k_proj<<<(N_NODES / 16) * (HF / 16) / 8, 256, 0, stream>>>(feat, W_fc, fs);

    // 2) attention logits per (node, head): one wave each
    k_elr<<<(N_NODES * HEADS + 7) / 8, 256, 0, stream>>>(fs, attn_l, attn_r, el, er);

    // 3) init output (residual + bias) and softmax accumulators
    k_init<<<(N_NODES * HF + 255) / 256, 256, 0, stream>>>(feat, bias, out, emax, esum);

    // 4) edge logits + per-dst max
    k_edge_logit<<<(N_EDGES * HEADS + 255) / 256, 256, 0, stream>>>(src, dst, el, er, e, emax);

    // 5) edge exp + per-dst sum
    k_edge_exp<<<(N_EDGES * HEADS + 255) / 256, 256, 0, stream>>>(dst, e, emax, esum);

    // 6) normalized scatter-add into output
    const unsigned int nthr = (unsigned int)N_EDGES * HEADS * 32u;
    k_scatter<<<(nthr + 255u) / 256u, 256, 0, stream>>>(src, dst, fs, e, esum, out);
}